// ListNetRankingLoss_45689862095269
// MI455X (gfx1250) — compile-verified
//
#include <hip/hip_runtime.h>
#include <hip/hip_bf16.h>

#define NDATES 4096
#define EXP_M5 0.006737946999085467f /* exp(-5) */

typedef __attribute__((ext_vector_type(2))) float v2f;
typedef __attribute__((ext_vector_type(8))) float v8f;

// ---------------- init: zero the 4096-entry per-date accumulators ----------
__global__ void ln_init(unsigned* __restrict__ maxbits, float* __restrict__ sum_e,
                        unsigned* __restrict__ cnt, unsigned* __restrict__ n1,
                        float* __restrict__ gce) {
  int t = blockIdx.x * blockDim.x + threadIdx.x;
  if (t < NDATES) {
    maxbits[t] = 0u;   // pred > 0 => float bits compare like uints, 0 == -inf here
    sum_e[t]   = 0.0f;
    cnt[t]     = 0u;
    n1[t]      = 0u;
    gce[t]     = 0.0f;
  }
}

// ---------------- pass A: per-date max(pred), count, count(label==1) -------
__global__ void ln_pass_a(const float2* __restrict__ scores,
                          const int* __restrict__ labels,
                          const int* __restrict__ dates,
                          unsigned* __restrict__ gmax, unsigned* __restrict__ gcnt,
                          unsigned* __restrict__ gn1, int n) {
  __shared__ unsigned smax[NDATES];
  __shared__ unsigned scnt[NDATES];
  __shared__ unsigned sn1[NDATES];
  for (int t = threadIdx.x; t < NDATES; t += blockDim.x) {
    smax[t] = 0u; scnt[t] = 0u; sn1[t] = 0u;
  }
  __syncthreads();

  int stride = gridDim.x * blockDim.x;
  for (int i = blockIdx.x * blockDim.x + threadIdx.x; i < n; i += stride) {
    __builtin_prefetch(&scores[i + stride], 0, 0);   // global_prefetch_b8
    float2 s   = scores[i];
    float pred = 1.0f / (1.0f + expf(s.x - s.y));    // softmax(scores)[:,1]
    int d      = dates[i];
    atomicMax(&smax[d], __float_as_uint(pred));      // ds_max_u32
    atomicAdd(&scnt[d], 1u);                         // ds_add_u32
    atomicAdd(&sn1[d], (unsigned)labels[i]);
  }
  __syncthreads();

  for (int t = threadIdx.x; t < NDATES; t += blockDim.x) {
    unsigned c = scnt[t];
    if (c) {
      atomicMax(&gmax[t], smax[t]);                  // global_atomic_max_u32
      atomicAdd(&gcnt[t], c);
      unsigned k = sn1[t];
      if (k) atomicAdd(&gn1[t], k);
    }
  }
}

// ---------------- pass B: per-date sum of exp(pred - max) ------------------
__global__ void ln_pass_b(const float2* __restrict__ scores,
                          const int* __restrict__ dates,
                          const unsigned* __restrict__ gmax,
                          float* __restrict__ gsum, int n) {
  __shared__ float smaxf[NDATES];
  __shared__ float ssum[NDATES];
  for (int t = threadIdx.x; t < NDATES; t += blockDim.x) {
    smaxf[t] = __uint_as_float(gmax[t]);
    ssum[t]  = 0.0f;
  }
  __syncthreads();

  int stride = gridDim.x * blockDim.x;
  for (int i = blockIdx.x * blockDim.x + threadIdx.x; i < n; i += stride) {
    __builtin_prefetch(&scores[i + stride], 0, 0);
    float2 s   = scores[i];
    float pred = 1.0f / (1.0f + expf(s.x - s.y));
    int d      = dates[i];
    atomicAdd(&ssum[d], expf(pred - smaxf[d]));      // ds_add_f32
  }
  __syncthreads();

  for (int t = threadIdx.x; t < NDATES; t += blockDim.x) {
    float v = ssum[t];
    if (v != 0.0f) atomicAdd(&gsum[t], v);           // global_atomic_add_f32
  }
}

// ---------------- prep: per-date softmax constants -------------------------
__global__ void ln_prep(const float* __restrict__ gsum, const unsigned* __restrict__ gcnt,
                        const unsigned* __restrict__ gn1, float* __restrict__ invden,
                        float* __restrict__ td0, float* __restrict__ td1) {
  int t = blockIdx.x * blockDim.x + threadIdx.x;
  if (t >= NDATES) return;
  float s   = gsum[t];
  invden[t] = (s > 0.0f) ? (1.0f / s) : 0.0f;
  unsigned c = gcnt[t], k = gn1[t];
  if (k) {                       // true-dist softmax max is 5
    float dn = (float)k + (float)(c - k) * EXP_M5;
    float inv = 1.0f / dn;
    td1[t] = inv;
    td0[t] = EXP_M5 * inv;
  } else {                       // all labels 0: uniform 1/c
    td1[t] = 0.0f;
    td0[t] = c ? (1.0f / (float)c) : 0.0f;
  }
}

// ---------------- pass C: per-date cross-entropy accumulation --------------
__global__ void ln_pass_c(const float2* __restrict__ scores,
                          const int* __restrict__ labels,
                          const int* __restrict__ dates,
                          const unsigned* __restrict__ gmax,
                          const float* __restrict__ invden,
                          const float* __restrict__ td0,
                          const float* __restrict__ td1,
                          float* __restrict__ gce, int n) {
  __shared__ float smaxf[NDATES];
  __shared__ float sinv[NDATES];
  __shared__ float sce[NDATES];
  for (int t = threadIdx.x; t < NDATES; t += blockDim.x) {
    smaxf[t] = __uint_as_float(gmax[t]);
    sinv[t]  = invden[t];
    sce[t]   = 0.0f;
  }
  __syncthreads();

  int stride = gridDim.x * blockDim.x;
  for (int i = blockIdx.x * blockDim.x + threadIdx.x; i < n; i += stride) {
    __builtin_prefetch(&scores[i + stride], 0, 0);
    float2 s   = scores[i];
    float pred = 1.0f / (1.0f + expf(s.x - s.y));
    int d      = dates[i];
    float pd   = expf(pred - smaxf[d]) * sinv[d];    // pred_dist
    float td   = labels[i] ? td1[d] : td0[d];        // true_dist (L2-resident tables)
    atomicAdd(&sce[d], -td * logf(pd + 1e-8f));
  }
  __syncthreads();

  for (int t = threadIdx.x; t < NDATES; t += blockDim.x) {
    float v = sce[t];
    if (v != 0.0f) atomicAdd(&gce[t], v);
  }
}

// ---------------- final: masked sum + valid count via WMMA on one wave -----
// A = ones(16x4). D[m][n] = sum_k B[k][n]  => every row of D holds the column
// sums; accumulating C over 64 chunks and summing one row gives the total of
// all 4096 elements, entirely in fp32 on the matrix pipe. Loads are
// unconditional + multiplicative masking so EXEC stays all-ones (no
// saveexec branches around the WMMAs).
__global__ void ln_final(const float2* __restrict__ gce2,
                         const uint2* __restrict__ gcnt2,
                         float* __restrict__ out) {
  int lane = threadIdx.x;            // one wave32, EXEC all ones throughout
  v2f a; a.x = 1.0f; a.y = 1.0f;     // A-matrix: all ones (16x4 f32, 2 VGPRs)
  v8f accC = {};                     // masked-ce accumulator
  v8f accV = {};                     // valid-count accumulator

  for (int base = 0; base < NDATES / 2; base += 32) {
    float2 c = gce2[base + lane];    // global_load_b64, all lanes
    uint2  k = gcnt2[base + lane];   // global_load_b64, all lanes
    float m0 = (k.x >= 2u) ? 1.0f : 0.0f;   // v_cndmask
    float m1 = (k.y >= 2u) ? 1.0f : 0.0f;
    v2f b;  b.x  = c.x * m0;  b.y  = c.y * m1;
    v2f bv; bv.x = m0;        bv.y = m1;
    accC = __builtin_amdgcn_wmma_f32_16x16x4_f32(false, a, false, b,
                                                 (short)0, accC, false, false);
    accV = __builtin_amdgcn_wmma_f32_16x16x4_f32(false, a, false, bv,
                                                 (short)0, accV, false, false);
  }

  // D row (VGPR0): lanes 0..15 hold N=0..15 column sums (lanes 16..31 hold the
  // identical M=8 row). XOR-reduce within each 16-lane half.
  float sc = accC[0];
  float sv = accV[0];
  for (int off = 8; off >= 1; off >>= 1) {
    sc += __shfl_xor(sc, off, 32);
    sv += __shfl_xor(sv, off, 32);
  }
  if (lane == 0) out[0] = sc / fmaxf(sv, 1.0f);
}

extern "C" void kernel_launch(void* const* d_in, const int* in_sizes, int n_in,
                              void* d_out, int out_size, void* d_ws, size_t ws_size,
                              hipStream_t stream) {
  const float* scores = (const float*)d_in[0];   // [B,2] f32
  const int*   labels = (const int*)d_in[1];     // [B] i32 (jax default x64-off)
  const int*   dates  = (const int*)d_in[2];     // [B] i32 in [0,4096)
  int n = in_sizes[1];

  float* ws = (float*)d_ws;                      // 8*16KB = 128KB workspace
  unsigned* maxbits = (unsigned*)ws;             // per-date max(pred) as bits
  float*    sum_e   = ws + 1 * NDATES;
  unsigned* cnt     = (unsigned*)(ws + 2 * NDATES);
  unsigned* n1      = (unsigned*)(ws + 3 * NDATES);
  float*    invden  = ws + 4 * NDATES;
  float*    td0     = ws + 5 * NDATES;
  float*    td1     = ws + 6 * NDATES;
  float*    gce     = ws + 7 * NDATES;

  const int blocks = 512;                        // 64 rows/thread; 2M merge atomics total
  ln_init<<<NDATES / 256, 256, 0, stream>>>(maxbits, sum_e, cnt, n1, gce);
  ln_pass_a<<<blocks, 256, 0, stream>>>((const float2*)scores, labels, dates,
                                        maxbits, cnt, n1, n);
  ln_pass_b<<<blocks, 256, 0, stream>>>((const float2*)scores, dates,
                                        maxbits, sum_e, n);
  ln_prep<<<NDATES / 256, 256, 0, stream>>>(sum_e, cnt, n1, invden, td0, td1);
  ln_pass_c<<<blocks, 256, 0, stream>>>((const float2*)scores, labels, dates,
                                        maxbits, invden, td0, td1, gce, n);
  ln_final<<<1, 32, 0, stream>>>((const float2*)gce, (const uint2*)cnt,
                                 (float*)d_out);
}